// WpSumAgg_48473000903091
// MI455X (gfx1250) — compile-verified
//
#include <hip/hip_runtime.h>

// Problem constants (from reference): B=1, T=8, C=64, H=W=256, HD=4, CH=16,
// PS=7, Q=32768, K=10 (of K_TOT=15).
#define PSZ      7
#define KNUM     10
#define KTOT     15
#define NPOS     49      // 7*7
#define NELEM    3136    // 4 heads * 16 ch * 49 pos
#define THREADS  256
#define PER_T    13      // ceil(3136/256)
#define NPAD     (PER_T * THREADS)   // 3328, padded LDS slab
#define NBUF     3                   // triple buffer -> 1 barrier per k

#if defined(__HIP_DEVICE_COMPILE__)
  #if __has_builtin(__builtin_amdgcn_global_load_async_to_lds_b32)
    #define HAVE_ASYNC_BUILTIN 1
  #endif
  #if __has_builtin(__builtin_amdgcn_s_wait_asynccnt)
    #define HAVE_WAITASYNC_BUILTIN 1
  #endif
#endif

typedef __attribute__((address_space(1))) int g_int;
typedef __attribute__((address_space(3))) int l_int;

__device__ __forceinline__ void async_copy_dw(const float* gsrc, float* ldst) {
#if defined(HAVE_ASYNC_BUILTIN)
  __builtin_amdgcn_global_load_async_to_lds_b32(
      (g_int*)(void*)gsrc,            // AS1 (global) int*
      (l_int*)(void*)ldst,            // AS3 (LDS) int*
      /*imm offset*/0, /*cpol*/0);
#elif defined(__HIP_DEVICE_COMPILE__)
  unsigned lofs = (unsigned)(unsigned long long)ldst;
  asm volatile("global_load_async_to_lds_b32 %0, %1, off"
               :: "v"(lofs), "v"(gsrc) : "memory");
#else
  (void)gsrc; (void)ldst;
#endif
}

template <int N>
__device__ __forceinline__ void wait_asynccnt_le() {
#if defined(HAVE_WAITASYNC_BUILTIN)
  __builtin_amdgcn_s_wait_asynccnt(N);
#elif defined(__HIP_DEVICE_COMPILE__)
  asm volatile("s_wait_asynccnt %0" :: "i"(N) : "memory");
#endif
}

__global__ __launch_bounds__(THREADS)
void WpSumAgg_48473000903091_kernel(const float* __restrict__ vid,
                                    const float* __restrict__ dists,
                                    const int*   __restrict__ inds,
                                    float*       __restrict__ out) {
  __shared__ float wLds[4 * KNUM];
  __shared__ int   baseLds[4 * KNUM];          // dword offset of patch origin
  __shared__ float pbuf[NBUF][NPAD];           // triple-buffered patch slabs

  const int tid = threadIdx.x;
  const int q   = blockIdx.x;                  // one block per query

  // ---- stage weights + precomputed base offsets for all (hd,k) ----
  if (tid < 4 * KNUM) {
    const int hd = tid / KNUM;
    const int k  = tid % KNUM;
    const long long r = ((long long)hd * 32768 + q) * KTOT + k;
    wLds[tid] = dists[r];
    const int* ip = inds + r * 3;
    // dword offset: ((t*64 + hd*16) * 256 + hi) * 256 + wi
    baseLds[tid] = ((ip[0] * 64 + hd * 16) << 16) + (ip[1] << 8) + ip[2];
  }
  __syncthreads();

  // ---- per-thread staging decomposition (independent of k) ----
  // element e = hdch*49 + (ph*7+pw); lanes take consecutive e -> consecutive
  // pw -> row-contiguous global addresses (coalesced 28B rows).
  int sOff[PER_T];
  int sHd[PER_T];
#pragma unroll
  for (int i = 0; i < PER_T; ++i) {
    int e  = tid + i * THREADS;
    int ee = (e < NELEM) ? e : (NELEM - 1);
    int hdch = ee / NPOS;
    int p    = ee - hdch * NPOS;
    int ph   = p / PSZ;
    int pw   = p - ph * PSZ;
    sHd[i]  = hdch >> 4;
    sOff[i] = ((hdch & 15) << 16) + (ph << 8) + pw;  // ch*65536 + ph*256 + pw
  }

  const int hdch = tid & 63;   // output channel (hd*16+ch), coalesced stores
  const int pg   = tid >> 6;   // position group: p in {pg, pg+4, ...}

  float acc[PER_T];
#pragma unroll
  for (int i = 0; i < PER_T; ++i) acc[i] = 0.0f;

  // ---- prime the pipeline: stage neighbor k=0 into pbuf[0] ----
#pragma unroll
  for (int i = 0; i < PER_T; ++i) {
    const int off = baseLds[sHd[i] * KNUM + 0] + sOff[i];
    async_copy_dw(vid + off, &pbuf[0][tid + i * THREADS]);
  }

  for (int k = 0; k < KNUM; ++k) {
    // stage batch k+1 into slab (k+1)%3: its previous reader was compute(k-2),
    // which all waves finished before the pre-compute barrier of iter k-1.
    if (k + 1 < KNUM) {
      float* dst = pbuf[(k + 1) % NBUF];
#pragma unroll
      for (int i = 0; i < PER_T; ++i) {
        const int off = baseLds[sHd[i] * KNUM + (k + 1)] + sOff[i];
        async_copy_dw(vid + off, &dst[tid + i * THREADS]);
      }
      wait_asynccnt_le<PER_T>();   // async loads retire in order: batch k done
    } else {
      wait_asynccnt_le<0>();
    }
    __syncthreads();               // make batch-k LDS writes block-visible

    const float w = wLds[(hdch >> 4) * KNUM + k];
    const float* src = &pbuf[k % NBUF][hdch * NPOS];  // stride 49: conflict-free
#pragma unroll
    for (int i = 0; i < PER_T; ++i) {
      const int p = pg + i * 4;
      if (p < NPOS) acc[i] += w * src[p];
    }
  }

  // ---- coalesced, NON-TEMPORAL output: 411 MB streaming writes must not
  // evict the L2-resident 134 MB vid that all 32768 blocks keep gathering ----
  float* op = out + (long long)q * NPOS * 64 + hdch;
#pragma unroll
  for (int i = 0; i < PER_T; ++i) {
    const int p = pg + i * 4;
    if (p < NPOS) __builtin_nontemporal_store(acc[i], op + (long long)p * 64);
  }
}

extern "C" void kernel_launch(void* const* d_in, const int* in_sizes, int n_in,
                              void* d_out, int out_size, void* d_ws, size_t ws_size,
                              hipStream_t stream) {
  const float* vid   = (const float*)d_in[0];
  const float* dists = (const float*)d_in[1];
  const int*   inds  = (const int*)d_in[2];
  float* out = (float*)d_out;
  (void)in_sizes; (void)n_in; (void)out_size; (void)d_ws; (void)ws_size;

  dim3 grid(32768);   // Q queries, one block each
  dim3 block(THREADS);
  hipLaunchKernelGGL(WpSumAgg_48473000903091_kernel, grid, block, 0, stream,
                     vid, dists, inds, out);
}